// RollScheduler_63273458204913
// MI455X (gfx1250) — compile-verified
//
#include <hip/hip_runtime.h>

// out[row, k] = (k >= idx[row]) ? clips[row, k - idx[row]] : 0
// rows = B*C = 1024, N = 32768 (derived from in_sizes at launch).
// Pure bandwidth problem: 256 MiB traffic -> ~11 us floor at 23.3 TB/s.
// Bulk chunks ride the gfx1250 async global<->LDS data mover (ASYNCcnt);
// boundary/zero chunks use a direct predicated float4 path.

#if defined(__gfx1250__) &&                                                    \
    __has_builtin(__builtin_amdgcn_global_load_async_to_lds_b32) &&            \
    __has_builtin(__builtin_amdgcn_global_store_async_from_lds_b128)
#define USE_GFX1250_ASYNC 1
#else
#define USE_GFX1250_ASYNC 0
#endif

#if USE_GFX1250_ASYNC
typedef int v4i __attribute__((ext_vector_type(4)));
typedef __attribute__((address_space(1))) int  as1_i32;
typedef __attribute__((address_space(3))) int  as3_i32;
typedef __attribute__((address_space(1))) v4i  as1_v4i;
typedef __attribute__((address_space(3))) v4i  as3_v4i;

__device__ __forceinline__ void wait_asynccnt0() {
#if __has_builtin(__builtin_amdgcn_s_wait_asynccnt)
  __builtin_amdgcn_s_wait_asynccnt(0);
#else
  asm volatile("s_wait_asynccnt 0x0" ::: "memory");
#endif
}
#endif

__global__ __launch_bounds__(256) void roll_place_kernel(
    const float* __restrict__ x, const float* __restrict__ clips,
    float* __restrict__ out, int N) {
  const int row = blockIdx.y;                       // uniform -> scalar loads
  const int idx = (int)(x[row] * (float)N);         // matches jnp trunc cast
  const long long base = (long long)row * (long long)N;
  const float* __restrict__ src = clips + base;
  float* __restrict__ dst = out + base;

  const int k0 = (int)(blockIdx.x * blockDim.x + threadIdx.x) * 4;
  if (k0 >= N) return;
  const int s = k0 - idx;                           // source index of element 0

#if USE_GFX1250_ASYNC
  // Bulk path: fully in-range 16B chunks go through the async LDS data mover.
  // Source side is only 4B-aligned (shift by idx), so loads are async B32;
  // destination side is 16B-aligned, so the store is a single async B128.
  __shared__ float lbuf[256 * 4];
  float* myl = &lbuf[threadIdx.x * 4];
  const bool full = (s >= 0);
  if (full) {
    __builtin_amdgcn_global_load_async_to_lds_b32((as1_i32*)(src + s + 0),
                                                  (as3_i32*)(myl + 0), 0, 0);
    __builtin_amdgcn_global_load_async_to_lds_b32((as1_i32*)(src + s + 1),
                                                  (as3_i32*)(myl + 1), 0, 0);
    __builtin_amdgcn_global_load_async_to_lds_b32((as1_i32*)(src + s + 2),
                                                  (as3_i32*)(myl + 2), 0, 0);
    __builtin_amdgcn_global_load_async_to_lds_b32((as1_i32*)(src + s + 3),
                                                  (as3_i32*)(myl + 3), 0, 0);
  } else {
    // Zero region and the single straddling chunk: direct VGPR path.
    float4 v;
    v.x = (s + 0 >= 0) ? src[s + 0] : 0.0f;
    v.y = (s + 1 >= 0) ? src[s + 1] : 0.0f;
    v.z = (s + 2 >= 0) ? src[s + 2] : 0.0f;
    v.w = (s + 3 >= 0) ? src[s + 3] : 0.0f;
    *(float4*)(dst + k0) = v;
  }
  wait_asynccnt0();  // async loads complete before async store reads LDS
  if (full) {
    __builtin_amdgcn_global_store_async_from_lds_b128((as1_v4i*)(dst + k0),
                                                      (as3_v4i*)myl, 0, 0);
  }
  // S_ENDPGM performs an implicit wait-idle, covering the outstanding store.
#else
  // Fallback: direct predicated copy. Stores are 128-bit aligned; source
  // loads are dword accesses that still coalesce into full cachelines.
  float4 v;
  v.x = (s + 0 >= 0) ? src[s + 0] : 0.0f;
  v.y = (s + 1 >= 0) ? src[s + 1] : 0.0f;
  v.z = (s + 2 >= 0) ? src[s + 2] : 0.0f;
  v.w = (s + 3 >= 0) ? src[s + 3] : 0.0f;
  *(float4*)(dst + k0) = v;
#endif
}

extern "C" void kernel_launch(void* const* d_in, const int* in_sizes, int n_in,
                              void* d_out, int out_size, void* d_ws,
                              size_t ws_size, hipStream_t stream) {
  (void)n_in; (void)out_size; (void)d_ws; (void)ws_size;
  const float* x     = (const float*)d_in[0];  // (B, C) positions in [0,1)
  const float* clips = (const float*)d_in[1];  // (B, C, N)
  // d_in[2] ("actual") is unused in the forward pass.
  float* out = (float*)d_out;                  // (B, C, N)

  const int rows = in_sizes[0];                // B*C
  const int N    = in_sizes[1] / rows;         // 32768

  dim3 block(256);
  dim3 grid((unsigned)((N / 4 + 255) / 256), (unsigned)rows);
  hipLaunchKernelGGL(roll_place_kernel, grid, block, 0, stream,
                     x, clips, out, N);
}